// BatchedLoRA_90013924590282
// MI455X (gfx1250) — compile-verified
//
#include <hip/hip_runtime.h>

typedef __attribute__((ext_vector_type(2))) float v2f;
typedef __attribute__((ext_vector_type(4))) float v4f;
typedef __attribute__((ext_vector_type(8))) float v8f;
typedef __attribute__((ext_vector_type(4))) unsigned int tdm_v4u;
typedef __attribute__((ext_vector_type(8))) int tdm_v8i;
typedef __attribute__((ext_vector_type(4))) int tdm_v4i;

#define T_TOT   8192
#define D_IN    2048
#define D_OUT   2048
#define RANK    64
#define NEXP    8
#define TTILE   128        // rows of x per workgroup (8 waves x 16-row strips)
#define DCHUNK  128        // K-chunk of A_n staged in LDS
#define OCHUNK  128        // D_OUT chunk of B_n staged in LDS
#define WSTRIDE 132        // padded LDS row stride for A chunk; 132%64==4 -> conflict-free
#define ASTRIDE 68         // padded LDS row stride for a_down tile / B chunk; 68%64==4
#define NCHA    (D_IN / DCHUNK)     // 16 A-chunks
#define NCHB    (D_OUT / OCHUNK)    // 16 B-chunks
#define NCH     (NCHA + NCHB)       // 32 chunks streamed through LDS
#define BUFSZ   8704       // floats per buffer: max(64*132=8448, 128*68=8704)
#define SCALING 2.0f       // ALPHA / RANK = 128/64

#if __has_builtin(__builtin_amdgcn_tensor_load_to_lds) && \
    __has_builtin(__builtin_amdgcn_s_wait_tensorcnt)
#define USE_TDM 1
#else
#define USE_TDM 0
#endif

__device__ __forceinline__ v8f wmma_f32(v2f a, v2f b, v8f c) {
  // V_WMMA_F32_16X16X4_F32 : D = A(16x4) * B(4x16) + C(16x16)
  return __builtin_amdgcn_wmma_f32_16x16x4_f32(false, a, false, b, (short)0, c,
                                               false, false);
}

#if USE_TDM
// 2-D TDM load: tile [tile1 rows][tile0 elems] of f32, row pitch `pitch0` elems,
// into LDS at byte offset ldsOff with pad_amount(+1) dwords inserted every
// 2^(padInt+1) dwords (reproduces the padded LDS row strides).
__device__ __forceinline__ void tdm_load_2d(const float* gsrc, unsigned ldsOff,
                                            unsigned tile0, unsigned tile1,
                                            unsigned pitch0,
                                            unsigned padInt, unsigned padAmt) {
  unsigned long long ga = (unsigned long long)(size_t)gsrc;
  tdm_v4u g0;
  g0.x = 1u;                                   // count=1 (valid), user mode
  g0.y = ldsOff;                               // lds_addr (bytes)
  g0.z = (unsigned)ga;                         // global_addr[31:0]
  g0.w = (unsigned)(ga >> 32) | (2u << 30);    // global_addr[56:32] | type=2
  tdm_v8i g1;
  g1[0] = (int)((2u << 16)            // data_size = 4B
              | (1u << 20)            // pad_enable
              | (padInt << 22) | (padAmt << 25));
  g1[1] = (int)((tile0 & 0xffffu) << 16);                    // tensor_dim0[15:0]
  g1[2] = (int)((tile0 >> 16) | ((tile1 & 0xffffu) << 16));  // td0 hi | tensor_dim1 lo
  g1[3] = (int)((tile1 >> 16) | (tile0 << 16));              // td1 hi | tile_dim0
  g1[4] = (int)(tile1 & 0xffffu);                            // tile_dim1 ; tile_dim2=0
  g1[5] = (int)pitch0;                                       // tensor_dim0_stride lo32
  g1[6] = 0;                                                 // stride0 hi | stride1 lo
  g1[7] = 0;
  tdm_v4i z4 = {};
#if defined(__clang_major__) && (__clang_major__ >= 23)
  tdm_v8i z8 = {};
  __builtin_amdgcn_tensor_load_to_lds(g0, g1, z4, z4, z8, 0);
#else
  __builtin_amdgcn_tensor_load_to_lds(g0, g1, z4, z4, 0);
#endif
}
#endif

__global__ __launch_bounds__(256, 1)
void lora_fused_kernel(const float* __restrict__ x,
                       const float* __restrict__ Aw,
                       const float* __restrict__ Bw,
                       float* __restrict__ out) {
  __shared__ float lds_buf[2 * BUFSZ];        // double-buffered weight chunks
  __shared__ float lds_a[TTILE * ASTRIDE];    // a_down tile, layout [t_local][r]

  const int tid  = threadIdx.x;
  const int lane = tid & 31;
  const int wave = tid >> 5;
  const int lm   = lane & 15;      // row (A-frag M) / col (B-frag N) within 16x16
  const int half = lane >> 4;      // selects K pair in A/B fragments
  const int kofs = half * 2;

  const int bid = blockIdx.x;
  const int n   = bid & (NEXP - 1);          // expert; neighbors share x tile in L2
  const int t0  = (bid >> 3) * TTILE;

  v8f acc[4] = {};        // stage-1: D(r x t): 4 tiles, r0 = rt*16, t = wave strip
  v2f afrag[16] = {};     // stage-2: a_down^T strip as 16 B-fragments (K=64)

  // x as B-matrix fragment: lane -> col t = t0+wave*16+lm, k = d + kofs + {0,1}
  const float* xbase = x + (size_t)(t0 + wave * 16 + lm) * D_IN + kofs;

#if USE_TDM
  auto issue_chunk = [&](int c) {
    unsigned ldsOff = (unsigned)(size_t)(void*)(lds_buf + (c & 1) * BUFSZ);
    if (c < NCHA) {
      // A_n chunk: 64 rows x 128 elems, pitch 2048; pad 4 dw every 128 dw -> stride 132
      tdm_load_2d(Aw + (size_t)n * RANK * D_IN + c * DCHUNK, ldsOff,
                  DCHUNK, RANK, D_IN, /*padInt=*/6, /*padAmt=*/3);
    } else {
      // B_n chunk: 128 rows x 64 elems, pitch 64; pad 4 dw every 64 dw -> stride 68
      tdm_load_2d(Bw + (size_t)(n * D_OUT + (c - NCHA) * OCHUNK) * RANK, ldsOff,
                  RANK, OCHUNK, RANK, /*padInt=*/5, /*padAmt=*/3);
    }
  };
  if (wave == 0) issue_chunk(0);   // prologue: start DMA of first chunk
#endif

  for (int c = 0; c < NCH; ++c) {
    float* buf = lds_buf + (c & 1) * BUFSZ;

#if USE_TDM
    if (wave == 0) {
      if (c + 1 < NCH) {
        issue_chunk(c + 1);                          // prefetch next chunk (other buffer)
        __builtin_amdgcn_s_wait_tensorcnt((short)1); // in-order TDM => chunk c landed
      } else {
        __builtin_amdgcn_s_wait_tensorcnt((short)0);
      }
    }
#else
    // fallback: cooperative vector loads into the padded LDS layout
    if (c < NCHA) {
      const float* agl = Aw + (size_t)n * RANK * D_IN + c * DCHUNK;
      #pragma unroll
      for (int idx = tid; idx < RANK * (DCHUNK / 4); idx += 256) {
        int r = idx >> 5, q = idx & 31;            // DCHUNK/4 == 32 v4f per row
        *(v4f*)(buf + r * WSTRIDE + q * 4) =
            *(const v4f*)(agl + (size_t)r * D_IN + q * 4);
      }
    } else {
      const float* bgl = Bw + (size_t)(n * D_OUT + (c - NCHA) * OCHUNK) * RANK;
      #pragma unroll
      for (int idx = tid; idx < OCHUNK * (RANK / 4); idx += 256) {
        int o = idx >> 4, q = idx & 15;            // RANK/4 == 16 v4f per row
        *(v4f*)(buf + o * ASTRIDE + q * 4) =
            *(const v4f*)(bgl + (size_t)o * RANK + q * 4);
      }
    }
#endif
    __syncthreads();   // chunk c visible to all waves

    if (c < NCHA) {
      // === Stage 1: acc(r x t) += A_chunk(r x DCHUNK) * X_strip^T (DCHUNK x t) ===
      const float* wb = buf + lm * WSTRIDE + kofs;   // A-frag: row r = rt*16+lm
      const float* xb = xbase + c * DCHUNK;
      #pragma unroll 4
      for (int d = 0; d < DCHUNK; d += 4) {
        if ((d & 28) == 0 && c + 1 < NCHA)
          __builtin_prefetch(xb + d + DCHUNK, 0, 3);   // next chunk's x lines
        v2f xa = *(const v2f*)(xb + d);                // B-fragment (x columns)
        #pragma unroll
        for (int rt = 0; rt < 4; ++rt)
          acc[rt] = wmma_f32(*(const v2f*)(wb + rt * (16 * WSTRIDE) + d), xa,
                             acc[rt]);
      }
      if (c == NCHA - 1) {
        // spill D(r x t) -> lds_a[t][r]: lane holds col t=lm, 8 consecutive r
        float* arow = lds_a + (wave * 16 + lm) * ASTRIDE + half * 8;
        #pragma unroll
        for (int rt = 0; rt < 4; ++rt) {
          v4f s0 = {acc[rt][0], acc[rt][1], acc[rt][2], acc[rt][3]};
          v4f s1 = {acc[rt][4], acc[rt][5], acc[rt][6], acc[rt][7]};
          *(v4f*)(arow + rt * 16)     = s0;     // ds_store_b128
          *(v4f*)(arow + rt * 16 + 4) = s1;
        }
        // reload strip as B-fragments of a_down^T (K=64): lane t=lm, r=kt*4+kofs
        const float* ab = lds_a + (wave * 16 + lm) * ASTRIDE + kofs;
        #pragma unroll
        for (int kt = 0; kt < 16; ++kt)
          afrag[kt] = *(const v2f*)(ab + kt * 4);
      }
    } else {
      // === Stage 2: D(o x t) = B_chunk(o x 64) * a_down^T (64 x t) ===
      int oc = (c - NCHA) * OCHUNK;
      float* orow =
          out + ((size_t)(t0 + wave * 16 + lm) * NEXP + n) * D_OUT + half * 8;
      #pragma unroll 1
      for (int nt = 0; nt < OCHUNK / 16; nt += 2) {   // 2 tiles in flight
        v8f c0 = {}, c1 = {};
        const float* bb0 = buf + (nt * 16 + lm) * ASTRIDE + kofs;  // A-frag of B_n
        const float* bb1 = bb0 + 16 * ASTRIDE;
        #pragma unroll
        for (int kt = 0; kt < 16; ++kt) {
          v2f b0 = *(const v2f*)(bb0 + kt * 4);
          v2f b1 = *(const v2f*)(bb1 + kt * 4);
          c0 = wmma_f32(b0, afrag[kt], c0);
          c1 = wmma_f32(b1, afrag[kt], c1);
        }
        // lane holds row t=lm, 8 consecutive o values -> 2x b128 NT stores per tile
        int ob = oc + nt * 16;
        v4f s00 = {c0[0] * SCALING, c0[1] * SCALING, c0[2] * SCALING, c0[3] * SCALING};
        v4f s01 = {c0[4] * SCALING, c0[5] * SCALING, c0[6] * SCALING, c0[7] * SCALING};
        v4f s10 = {c1[0] * SCALING, c1[1] * SCALING, c1[2] * SCALING, c1[3] * SCALING};
        v4f s11 = {c1[4] * SCALING, c1[5] * SCALING, c1[6] * SCALING, c1[7] * SCALING};
        __builtin_nontemporal_store(s00, (v4f*)(orow + ob));
        __builtin_nontemporal_store(s01, (v4f*)(orow + ob + 4));
        __builtin_nontemporal_store(s10, (v4f*)(orow + ob + 16));
        __builtin_nontemporal_store(s11, (v4f*)(orow + ob + 20));
      }
    }
    __syncthreads();   // all waves done with buffer before it is refilled
  }
}

extern "C" void kernel_launch(void* const* d_in, const int* in_sizes, int n_in,
                              void* d_out, int out_size, void* d_ws, size_t ws_size,
                              hipStream_t stream) {
  const float* x  = (const float*)d_in[0];   // [8192, 2048] f32
  const float* Aw = (const float*)d_in[1];   // [8, 64, 2048] f32
  const float* Bw = (const float*)d_in[2];   // [8, 2048, 64] f32
  float* out = (float*)d_out;                // [8192, 8, 2048] f32

  dim3 grid(NEXP * (T_TOT / TTILE));         // 512 workgroups
  dim3 block(256);                           // 8 wave32
  lora_fused_kernel<<<grid, block, 0, stream>>>(x, Aw, Bw, out);
}